// VNDGCNN_75333726372155
// MI455X (gfx1250) — compile-verified
//
#include <hip/hip_runtime.h>
#include <hip/hip_bf16.h>
#include <math.h>

#define B_SZ   8
#define N_PTS  4096
#define C_CH   32
#define KNN    20
#define EPS_VN 1e-6f
#define EPS_BN 1e-5f

#define COLS_PER_PASS 2048           // float4 columns staged in LDS per TDM pass
#define ROWSTRIDE     20             // score-tile row stride (floats): 80B, 16B-aligned

typedef float v2f __attribute__((ext_vector_type(2)));
typedef float v8f __attribute__((ext_vector_type(8)));
typedef unsigned int v4u __attribute__((ext_vector_type(4)));
typedef int v4i __attribute__((ext_vector_type(4)));
typedef int v8i __attribute__((ext_vector_type(8)));

// ---------------------------------------------------------------------------
// Kernel 1: augment points -> float4 {x, y, z, -(x^2+y^2+z^2)}
// ---------------------------------------------------------------------------
__global__ void __launch_bounds__(256) k_aug(const float* __restrict__ x,
                                             float4* __restrict__ aug) {
    int i = blockIdx.x * 256 + threadIdx.x;          // over B*N
    if (i >= B_SZ * N_PTS) return;
    int b = i / N_PTS, n = i - b * N_PTS;
    const float* xb = x + (size_t)b * 3 * N_PTS;
    float px = xb[n], py = xb[N_PTS + n], pz = xb[2 * N_PTS + n];
    aug[i] = make_float4(px, py, pz, -(px * px + py * py + pz * pz));
}

// ---------------------------------------------------------------------------
// Kernel 2: kNN top-20.
//  - TDM (tensor_load_to_lds) stages 32KB column chunks into LDS per block.
//  - V_WMMA_F32_16X16X4_F32 scores 16 queries x 16 columns per instruction:
//    s = 2 q.c - |c|^2 (same per-row order as neg_dist; -|q|^2 row const drops).
//  - Per-row top-20 kept sorted in registers (static-index shift network).
// ---------------------------------------------------------------------------
__global__ void __launch_bounds__(128) k_knn(const float4* __restrict__ aug,
                                             int* __restrict__ knn_idx) {
    __shared__ float4 cols[COLS_PER_PASS];           // 32 KB TDM destination
    __shared__ float  tile[4][16 * ROWSTRIDE];       // per-wave 16x16 score tile

    const int lane = threadIdx.x & 31;
    const int w    = threadIdx.x >> 5;
    const int gw   = blockIdx.x * 4 + w;             // 2048 waves total
    const int b    = gw / (N_PTS / 16);              // same b for all 4 waves/block
    const int n0   = (gw - b * (N_PTS / 16)) * 16;
    const int m    = lane & 15;
    const int half = lane >> 4;
    const float4* ab = aug + (size_t)b * N_PTS;

    // A fragment (2 VGPRs): query row n0+m; lanes 0-15 hold K=0,1; 16-31 hold K=2,3
    float4 q = ab[n0 + m];
    v2f a;
    a.x = half ? 2.0f * q.z : 2.0f * q.x;
    a.y = half ? 1.0f       : 2.0f * q.y;

    // register-resident sorted top-20 (descending)
    float bs[KNN];
    int   bi[KNN];
#pragma unroll
    for (int t = 0; t < KNN; ++t) { bs[t] = -3.4e38f; bi[t] = 0; }

    auto insert = [&](float s, int c) {
        if (s > bs[KNN - 1]) {
            float ps = 3.4e38f; int pi = 0;          // sentinel bs[-1] = +inf
#pragma unroll
            for (int t = 0; t < KNN; ++t) {
                float cs = bs[t]; int ci = bi[t];
                if (s > cs) {
                    bool sh2 = (s > ps);             // already inserted above -> shift
                    bs[t] = sh2 ? ps : s;
                    bi[t] = sh2 ? pi : c;
                }
                ps = cs; pi = ci;
            }
        }
    };

    for (int pass = 0; pass < 2; ++pass) {
        __syncthreads();                             // prior pass fully consumed
        if (w == 0) {
            // ---- Tensor Data Mover: 32KB contiguous copy global -> LDS ----
            unsigned lds_off =
                (unsigned)(unsigned long long)(__attribute__((address_space(3))) char*)cols;
            unsigned long long ga =
                (unsigned long long)(const void*)(ab + pass * COLS_PER_PASS);
            const unsigned SZ = COLS_PER_PASS * 2;   // # of 8-byte elements = 4096
            v4u g0;
            g0[0] = 1u;                              // count=1 (valid user descriptor)
            g0[1] = lds_off;                         // D#.lds_addr
            g0[2] = (unsigned)(ga & 0xFFFFFFFFu);    // D#.global_addr[31:0]
            g0[3] = (unsigned)((ga >> 32) & 0x01FFFFFFu) | (2u << 30); // addr[56:32]|type=2
            v8i g1;
            g1[0] = (int)(3u << 16);                 // data_size=8B; wg_mask=0
            g1[1] = (int)((SZ & 0xFFFFu) << 16);     // tensor_dim0[15:0]
            g1[2] = (int)((SZ >> 16) | (1u << 16));  // tensor_dim0[31:16] | tensor_dim1=1
            g1[3] = (int)((SZ & 0xFFFFu) << 16);     // tile_dim0 = SZ
            g1[4] = 1;                               // tile_dim1=1, tile_dim2=0
            g1[5] = (int)SZ;                         // tensor_dim0_stride[31:0]
            g1[6] = 0;
            g1[7] = 0;
            v4i gz4 = {0, 0, 0, 0};
            v8i gz8 = {0, 0, 0, 0, 0, 0, 0, 0};
            __builtin_amdgcn_tensor_load_to_lds(g0, g1, gz4, gz4, gz8, 0);
            __builtin_amdgcn_s_wait_tensorcnt((short)0);
        }
        __syncthreads();                             // LDS chunk visible to all waves

        for (int ct2 = 0; ct2 < COLS_PER_PASS / 16; ++ct2) {
            const int cbase = pass * COLS_PER_PASS + ct2 * 16;
            // B fragment from LDS (ds_load_b128)
            float4 c4 = cols[ct2 * 16 + m];
            v2f bmat;
            bmat.x = half ? c4.z : c4.x;
            bmat.y = half ? c4.w : c4.y;

            v8f acc = {};
            acc = __builtin_amdgcn_wmma_f32_16x16x4_f32(
                false, a, false, bmat, (short)0, acc, false, false);

            // D layout: VGPR j -> row j (lanes 0-15) / j+8 (lanes 16-31); col = m
#pragma unroll
            for (int j = 0; j < 8; ++j)
                tile[w][(j + 8 * half) * ROWSTRIDE + m] = acc[j];
            __asm__ volatile("s_wait_dscnt 0x0" ::: "memory");

            if (half == 0) {                         // lane = row owner, b128 row reads
                const float4* rowp = (const float4*)&tile[w][m * ROWSTRIDE];
#pragma unroll
                for (int jj = 0; jj < 4; ++jj) {
                    float4 r = rowp[jj];
                    insert(r.x, cbase + jj * 4 + 0);
                    insert(r.y, cbase + jj * 4 + 1);
                    insert(r.z, cbase + jj * 4 + 2);
                    insert(r.w, cbase + jj * 4 + 3);
                }
            }
            // next-iter DS writes cannot pass these reads: DS in-order per wave
        }
    }

    if (half == 0) {
        int* op = knn_idx + ((size_t)(b * N_PTS + n0 + m)) * KNN;
#pragma unroll
        for (int t = 0; t < KNN; ++t) op[t] = bi[t];
    }
}

// ---------------------------------------------------------------------------
// Kernel 3: zero BN statistic accumulators (ws is not re-poisoned per call).
// ---------------------------------------------------------------------------
__global__ void k_zero(float* __restrict__ g) {
    if (threadIdx.x < 2 * C_CH) g[threadIdx.x] = 0.0f;
}

// ---------------------------------------------------------------------------
// Kernel 4: BN batch statistics. lane = channel; each wave walks 80 edges,
// accumulating sum / sumsq of ||p||+eps privately, then LDS + global atomics.
// ---------------------------------------------------------------------------
__global__ void __launch_bounds__(256) k_stats(const float4* __restrict__ aug,
                                               const int* __restrict__ knn_idx,
                                               const float* __restrict__ Wf,
                                               float* __restrict__ gstats) {
    __shared__ float ssum[C_CH], ssq[C_CH];
    const int tid = threadIdx.x;
    if (tid < C_CH) { ssum[tid] = 0.0f; ssq[tid] = 0.0f; }
    __syncthreads();

    const int lane = tid & 31, w = tid >> 5;
    const int o = lane;
    const float wf0 = Wf[o * 2 + 0], wf1 = Wf[o * 2 + 1];

    float acc = 0.0f, acc2 = 0.0f;
    const int base = (blockIdx.x * 8 + w) * 80;      // 8192 waves * 80 = 655360 edges
    for (int it = 0; it < 80; ++it) {
        int i = base + it;
        int b = i / (N_PTS * KNN);
        int r = i - b * (N_PTS * KNN);
        int n = r / KNN, k = r - n * KNN;
        int j = knn_idx[(size_t)(b * N_PTS + n) * KNN + k];
        float4 nb  = aug[b * N_PTS + j];
        float4 ctr = aug[b * N_PTS + n];
        float ex = nb.x - ctr.x, ey = nb.y - ctr.y, ez = nb.z - ctr.z;
        float px = wf0 * ex + wf1 * ctr.x;
        float py = wf0 * ey + wf1 * ctr.y;
        float pz = wf0 * ez + wf1 * ctr.z;
        float nrm = sqrtf(px * px + py * py + pz * pz) + EPS_VN;
        acc += nrm; acc2 += nrm * nrm;
    }
    atomicAdd(&ssum[o], acc);
    atomicAdd(&ssq[o], acc2);
    __syncthreads();
    if (tid < C_CH) {
        atomicAdd(&gstats[tid], ssum[tid]);
        atomicAdd(&gstats[C_CH + tid], ssq[tid]);
    }
}

// ---------------------------------------------------------------------------
// Kernel 5: fused VN linear + BN + leaky + learned-direction max pool.
// One wave per (b, n); lane = channel. xo (32x3) round-trips through LDS per
// neighbor so every lane forms dp = W_pool . xo from broadcast reads.
// ---------------------------------------------------------------------------
__global__ void __launch_bounds__(256) k_final(const float4* __restrict__ aug,
                                               const int* __restrict__ knn_idx,
                                               const float* __restrict__ Wf,
                                               const float* __restrict__ Wd,
                                               const float* __restrict__ gamma,
                                               const float* __restrict__ beta,
                                               const float* __restrict__ Wp,
                                               const float* __restrict__ gstats,
                                               float* __restrict__ out) {
    __shared__ float sh[8][C_CH * 4];
    const int lane = threadIdx.x & 31, w = threadIdx.x >> 5;
    const int gw = blockIdx.x * 8 + w;               // B*N waves
    const int b = gw / N_PTS, n = gw - b * N_PTS;
    const int o = lane;

    const float Mcnt = (float)(B_SZ * N_PTS * KNN);
    const float mean = gstats[o] / Mcnt;
    const float var  = gstats[C_CH + o] / Mcnt - mean * mean;
    const float grs  = gamma[o] * rsqrtf(var + EPS_BN);
    const float bt   = beta[o];
    const float wf0 = Wf[o * 2], wf1 = Wf[o * 2 + 1];
    const float wd0 = Wd[o * 2], wd1 = Wd[o * 2 + 1];

    float wp[C_CH];
#pragma unroll
    for (int i = 0; i < C_CH; ++i) wp[i] = Wp[o * C_CH + i];

    const float4 ctr = aug[b * N_PTS + n];
    const int* ip = knn_idx + ((size_t)(b * N_PTS + n)) * KNN;

    float best = -3.4e38f, bx = 0.0f, by = 0.0f, bz = 0.0f;
    for (int k = 0; k < KNN; ++k) {
        int j = ip[k];
        float4 nb = aug[b * N_PTS + j];
        float ex = nb.x - ctr.x, ey = nb.y - ctr.y, ez = nb.z - ctr.z;

        float px = wf0 * ex + wf1 * ctr.x;
        float py = wf0 * ey + wf1 * ctr.y;
        float pz = wf0 * ez + wf1 * ctr.z;
        float nrm = sqrtf(px * px + py * py + pz * pz) + EPS_VN;
        float nbn = grs * (nrm - mean) + bt;
        float sc  = nbn / nrm;
        px *= sc; py *= sc; pz *= sc;

        float dx = wd0 * ex + wd1 * ctr.x;
        float dy = wd0 * ey + wd1 * ctr.y;
        float dz = wd0 * ez + wd1 * ctr.z;
        float dot = px * dx + py * dy + pz * dz;
        float d2  = dx * dx + dy * dy + dz * dz + EPS_VN;
        float tf  = (dot >= 0.0f) ? 0.0f : 0.8f * (dot / d2);  // 0.2p+0.8(p-(dot/d2)d)
        float xx = px - tf * dx, xy = py - tf * dy, xz = pz - tf * dz;

        sh[w][o * 4 + 0] = xx; sh[w][o * 4 + 1] = xy; sh[w][o * 4 + 2] = xz;
        __asm__ volatile("s_wait_dscnt 0x0" ::: "memory");

        float dpx = 0.0f, dpy = 0.0f, dpz = 0.0f;
#pragma unroll
        for (int i = 0; i < C_CH; ++i) {
            float ww = wp[i];
            dpx += ww * sh[w][i * 4 + 0];
            dpy += ww * sh[w][i * 4 + 1];
            dpz += ww * sh[w][i * 4 + 2];
        }
        float score = xx * dpx + xy * dpy + xz * dpz;
        if (score > best) { best = score; bx = xx; by = xy; bz = xz; }
        // iteration k+1 LDS writes cannot pass iteration k reads (DS in-order)
    }

    size_t ob = ((size_t)(b * C_CH + o) * 3) * N_PTS + n;
    out[ob]             = bx;
    out[ob + N_PTS]     = by;
    out[ob + 2 * N_PTS] = bz;
}

// ---------------------------------------------------------------------------
extern "C" void kernel_launch(void* const* d_in, const int* in_sizes, int n_in,
                              void* d_out, int out_size, void* d_ws, size_t ws_size,
                              hipStream_t stream) {
    const float* x     = (const float*)d_in[0];   // [8,3,4096]
    const float* Wf    = (const float*)d_in[1];   // [32,2]
    const float* Wd    = (const float*)d_in[2];   // [32,2]
    const float* gamma = (const float*)d_in[3];   // [32]
    const float* beta  = (const float*)d_in[4];   // [32]
    const float* Wp    = (const float*)d_in[5];   // [32,32]
    float* out = (float*)d_out;                   // [8,32,3,4096]

    // workspace layout
    char* ws = (char*)d_ws;
    float4* aug   = (float4*)ws;                                    // 512 KiB
    int*    idxb  = (int*)(ws + (size_t)B_SZ * N_PTS * 16);         // 2.5 MiB
    float*  stats = (float*)(ws + (size_t)B_SZ * N_PTS * 16
                                + (size_t)B_SZ * N_PTS * KNN * 4);  // 64 floats

    k_aug  <<<(B_SZ * N_PTS + 255) / 256, 256, 0, stream>>>(x, aug);
    k_knn  <<<(B_SZ * (N_PTS / 16)) / 4, 128, 0, stream>>>(aug, idxb);
    k_zero <<<1, 64, 0, stream>>>(stats);
    k_stats<<<1024, 256, 0, stream>>>(aug, idxb, Wf, stats);
    k_final<<<(B_SZ * N_PTS) / 8, 256, 0, stream>>>(aug, idxb, Wf, Wd, gamma,
                                                    beta, Wp, stats, out);
    (void)in_sizes; (void)n_in; (void)out_size; (void)ws_size;
}